// GATLayer_7705171329327
// MI455X (gfx1250) — compile-verified
//
#include <hip/hip_runtime.h>

// ---------------------------------------------------------------------------
// GATv2 layer for MI455X (gfx1250, wave32).
//   x_l = x@Wl + bl ; x_r = x@Wr + br        (fp32 WMMA 16x16x4)
//   e   = att . leaky_relu(x_l[src]+x_r[dst])
//   segment softmax over dst, aggregate, +bias, +x (residual), LayerNorm, ELU
// ---------------------------------------------------------------------------

typedef float v2f __attribute__((ext_vector_type(2)));
typedef float v8f __attribute__((ext_vector_type(8)));

#define DIM        256      // IN_DIM == OUT_ALL
#define HEADS      8
#define HDIM       32
#define NEG_SLOPE  0.2f
#define LDS_STRIDE 264      // 256 + 8 pad: avoid LDS bank conflicts

// monotone float<->uint mapping for atomicMax on signed floats
__device__ __forceinline__ unsigned encf(float f) {
    unsigned u = __float_as_uint(f);
    return (u & 0x80000000u) ? ~u : (u | 0x80000000u);
}
__device__ __forceinline__ float decf(unsigned u) {
    return __uint_as_float((u & 0x80000000u) ? (u & 0x7fffffffu) : ~u);
}

// load 8 consecutive floats (32B aligned) into a scalar array via two b128s
__device__ __forceinline__ void load8(const float* __restrict__ p, float v[8]) {
    const float4 a = reinterpret_cast<const float4*>(p)[0];
    const float4 b = reinterpret_cast<const float4*>(p)[1];
    v[0] = a.x; v[1] = a.y; v[2] = a.z; v[3] = a.w;
    v[4] = b.x; v[5] = b.y; v[6] = b.z; v[7] = b.w;
}

// ---------------------------------------------------------------------------
// Kernel 1: dual GEMM. One block = 16-row tile of x staged in LDS; 8 waves,
// each wave owns one 16-col tile and accumulates BOTH Wl and Wr products with
// V_WMMA_F32_16X16X4_F32 (A fragment reused -> 2 WMMAs per K-step).
// Fragment layouts per CDNA5 ISA 7.12.2 (fp32, wave32).
// ---------------------------------------------------------------------------
__global__ __launch_bounds__(256)
void gat_gemm_xlxr(const float* __restrict__ x,
                   const float* __restrict__ Wl, const float* __restrict__ bl,
                   const float* __restrict__ Wr, const float* __restrict__ br,
                   float* __restrict__ xl, float* __restrict__ xr, int Nn)
{
    __shared__ float sx[16 * LDS_STRIDE];

    const int tid = threadIdx.x;
    // cooperative load of the 16 x 256 fp32 A-tile (each thread: 16 floats, b128s)
    {
        const int r  = tid >> 4;
        const int c0 = (tid & 15) * 16;
        int row = blockIdx.x * 16 + r;
        if (row >= Nn) row = Nn - 1;              // clamp (N is a multiple of 16 anyway)
        const float4* g = reinterpret_cast<const float4*>(x + (size_t)row * DIM + c0);
        float4* s = reinterpret_cast<float4*>(&sx[r * LDS_STRIDE + c0]);
        s[0] = g[0]; s[1] = g[1]; s[2] = g[2]; s[3] = g[3];
    }
    __syncthreads();

    const int w    = tid >> 5;          // wave id 0..7
    const int lane = tid & 31;
    const int half = lane >> 4;         // 0: lanes 0-15, 1: lanes 16-31
    const int m    = lane & 15;         // A row
    const int n    = lane & 15;         // B/C column within tile
    const int col0 = (blockIdx.y * 8 + w) * 16;

    v8f cl = {0.f,0.f,0.f,0.f,0.f,0.f,0.f,0.f};
    v8f cr = {0.f,0.f,0.f,0.f,0.f,0.f,0.f,0.f};

    #pragma unroll 4
    for (int k = 0; k < DIM; k += 4) {
        const int kb = k + 2 * half;    // this lane-half holds K = kb, kb+1
        v2f a;
        a.x = sx[m * LDS_STRIDE + kb];
        a.y = sx[m * LDS_STRIDE + kb + 1];
        v2f bL, bR;
        bL.x = Wl[(size_t)kb       * DIM + col0 + n];
        bL.y = Wl[(size_t)(kb + 1) * DIM + col0 + n];
        bR.x = Wr[(size_t)kb       * DIM + col0 + n];
        bR.y = Wr[(size_t)(kb + 1) * DIM + col0 + n];
        cl = __builtin_amdgcn_wmma_f32_16x16x4_f32(false, a, false, bL,
                                                   (short)0, cl, false, false);
        cr = __builtin_amdgcn_wmma_f32_16x16x4_f32(false, a, false, bR,
                                                   (short)0, cr, false, false);
    }

    const float biasl = bl[col0 + n];
    const float biasr = br[col0 + n];
    const int   row0  = blockIdx.x * 16;

    if (row0 + 16 <= Nn) {              // wave-uniform fast path: no per-row guards
        #pragma unroll
        for (int r = 0; r < 8; ++r) {   // VGPR r -> row r (lanes 0-15) / r+8 (lanes 16-31)
            const int row = row0 + r + 8 * half;
            xl[(size_t)row * DIM + col0 + n] = cl[r] + biasl;
            xr[(size_t)row * DIM + col0 + n] = cr[r] + biasr;
        }
    } else {
        #pragma unroll
        for (int r = 0; r < 8; ++r) {
            const int row = row0 + r + 8 * half;
            if (row < Nn) {
                xl[(size_t)row * DIM + col0 + n] = cl[r] + biasl;
                xr[(size_t)row * DIM + col0 + n] = cr[r] + biasr;
            }
        }
    }
}

// ---------------------------------------------------------------------------
// Kernel 2: per-edge GATv2 score + segment max.  One wave32 per edge; lane
// owns 8 contiguous dims (two b128 loads per side). 4 lanes form one head
// (HDIM=32); reduce with shfl_xor, lane leader stores score + atomicMax.
// ---------------------------------------------------------------------------
__global__ __launch_bounds__(256)
void gat_edge_score(const float* __restrict__ xl, const float* __restrict__ xr,
                    const int* __restrict__ ei, const float* __restrict__ att,
                    float* __restrict__ pe, unsigned* __restrict__ menc, int Ee)
{
    const int lane = threadIdx.x & 31;
    const int e = blockIdx.x * 8 + (threadIdx.x >> 5);
    if (e >= Ee) return;                            // wave-uniform exit

    const int src = ei[e];
    const int dst = ei[Ee + e];
    const int d0  = lane * 8;

    float lv[8], rv[8];
    load8(xl + (size_t)src * DIM + d0, lv);
    load8(xr + (size_t)dst * DIM + d0, rv);

    float part = 0.f;
    #pragma unroll
    for (int i = 0; i < 8; ++i) {
        float s = lv[i] + rv[i];
        s = (s > 0.f) ? s : NEG_SLOPE * s;
        part += s * att[d0 + i];                    // att flat [H][32] == flat [256]
    }
    part += __shfl_xor(part, 1, 32);
    part += __shfl_xor(part, 2, 32);                // sum over the 4 lanes of a head

    const int head = lane >> 2;
    if ((lane & 3) == 0) {
        pe[(size_t)e * HEADS + head] = part;
        atomicMax(&menc[(size_t)dst * HEADS + head], encf(part));
    }
}

// ---------------------------------------------------------------------------
// Kernel 3: p = exp(e - max) ; denom += p.  One thread per (edge, head).
// ---------------------------------------------------------------------------
__global__ __launch_bounds__(256)
void gat_exp_sum(const int* __restrict__ ei, const unsigned* __restrict__ menc,
                 float* __restrict__ pe, float* __restrict__ denom, int Ee)
{
    const int idx = blockIdx.x * blockDim.x + threadIdx.x;
    if (idx >= Ee * HEADS) return;
    const int e = idx >> 3, h = idx & 7;
    const int dst = ei[Ee + e];
    const float mv = decf(menc[(size_t)dst * HEADS + h]);
    const float p  = __expf(pe[idx] - mv);
    pe[idx] = p;
    atomicAdd(&denom[(size_t)dst * HEADS + h], p);
}

// ---------------------------------------------------------------------------
// Kernel 4: out[dst] += (p/denom[dst]) * x_l[src].  One wave per edge.
// ---------------------------------------------------------------------------
__global__ __launch_bounds__(256)
void gat_aggregate(const float* __restrict__ xl, const int* __restrict__ ei,
                   const float* __restrict__ pe, const float* __restrict__ denom,
                   float* __restrict__ out, int Ee)
{
    const int lane = threadIdx.x & 31;
    const int e = blockIdx.x * 8 + (threadIdx.x >> 5);
    if (e >= Ee) return;

    const int src  = ei[e];
    const int dst  = ei[Ee + e];
    const int d0   = lane * 8;
    const int head = lane >> 2;

    float dn = denom[(size_t)dst * HEADS + head];
    dn = (dn == 0.f) ? 1.f : dn;
    const float alpha = pe[(size_t)e * HEADS + head] / dn;

    float lv[8];
    load8(xl + (size_t)src * DIM + d0, lv);
    float* o = out + (size_t)dst * DIM + d0;
    #pragma unroll
    for (int i = 0; i < 8; ++i)
        atomicAdd(&o[i], alpha * lv[i]);
}

// ---------------------------------------------------------------------------
// Kernel 5: residual + LayerNorm + ELU.  One wave per node row (256 dims,
// lane owns 8); mean/var via wave32 shfl_xor tree.
// ---------------------------------------------------------------------------
__global__ __launch_bounds__(256)
void gat_norm_elu(const float* __restrict__ x, const float* __restrict__ bias,
                  const float* __restrict__ gamma, const float* __restrict__ beta,
                  float* __restrict__ out, int Nn)
{
    const int lane = threadIdx.x & 31;
    const int row = blockIdx.x * 8 + (threadIdx.x >> 5);
    if (row >= Nn) return;
    const int d0 = lane * 8;

    float z[8], s = 0.f, sq = 0.f;
    load8(out + (size_t)row * DIM + d0, z);
    #pragma unroll
    for (int i = 0; i < 8; ++i) {
        z[i] += bias[d0 + i] + x[(size_t)row * DIM + d0 + i];
        s  += z[i];
        sq += z[i] * z[i];
    }
    #pragma unroll
    for (int o = 1; o < 32; o <<= 1) {
        s  += __shfl_xor(s,  o, 32);
        sq += __shfl_xor(sq, o, 32);
    }
    const float mean = s * (1.0f / DIM);
    const float var  = sq * (1.0f / DIM) - mean * mean;
    const float rstd = rsqrtf(var + 1e-5f);
    #pragma unroll
    for (int i = 0; i < 8; ++i) {
        float zn = (z[i] - mean) * rstd * gamma[d0 + i] + beta[d0 + i];
        out[(size_t)row * DIM + d0 + i] = (zn > 0.f) ? zn : (__expf(zn) - 1.f);
    }
}

// ---------------------------------------------------------------------------
extern "C" void kernel_launch(void* const* d_in, const int* in_sizes, int n_in,
                              void* d_out, int out_size, void* d_ws, size_t ws_size,
                              hipStream_t stream)
{
    const float* x     = (const float*)d_in[0];
    const int*   ei    = (const int*)  d_in[1];
    const float* Wl    = (const float*)d_in[2];
    const float* bl    = (const float*)d_in[3];
    const float* Wr    = (const float*)d_in[4];
    const float* br    = (const float*)d_in[5];
    const float* att   = (const float*)d_in[6];
    const float* bias  = (const float*)d_in[7];
    const float* gamma = (const float*)d_in[8];
    const float* beta  = (const float*)d_in[9];
    float* out = (float*)d_out;

    const int Nn = in_sizes[0] / DIM;
    const int Ee = in_sizes[1] / 2;

    // workspace carve-out (fp32 slots): ~237 MB total
    float*    xl    = (float*)d_ws;
    float*    xr    = xl + (size_t)Nn * DIM;
    float*    pe    = xr + (size_t)Nn * DIM;
    float*    denom = pe + (size_t)Ee * HEADS;
    unsigned* menc  = (unsigned*)(denom + (size_t)Nn * HEADS);

    // per-call init (graph-capture-safe memset nodes)
    hipMemsetAsync(out,   0, (size_t)Nn * DIM   * sizeof(float), stream);
    hipMemsetAsync(denom, 0, (size_t)Nn * HEADS * sizeof(float), stream);
    hipMemsetAsync(menc,  0, (size_t)Nn * HEADS * sizeof(unsigned), stream); // 0 == -inf in encf order

    dim3 blk(256);
    dim3 gGemm((Nn + 15) / 16, 2);          // y: 2 * 8 waves = 16 col tiles of 16
    gat_gemm_xlxr<<<gGemm, blk, 0, stream>>>(x, Wl, bl, Wr, br, xl, xr, Nn);

    dim3 gEdge((Ee + 7) / 8);
    gat_edge_score<<<gEdge, blk, 0, stream>>>(xl, xr, ei, att, pe, menc, Ee);

    dim3 gExp((Ee * HEADS + 255) / 256);
    gat_exp_sum<<<gExp, blk, 0, stream>>>(ei, menc, pe, denom, Ee);

    gat_aggregate<<<gEdge, blk, 0, stream>>>(xl, ei, pe, denom, out, Ee);

    dim3 gNorm((Nn + 7) / 8);
    gat_norm_elu<<<gNorm, blk, 0, stream>>>(x, bias, gamma, beta, out, Nn);
}